// MemoryWithUsage_16999480558224
// MI455X (gfx1250) — compile-verified
//
#include <hip/hip_runtime.h>
#include <hip/hip_bf16.h>

typedef __attribute__((ext_vector_type(2))) float v2f;
typedef __attribute__((ext_vector_type(8))) float v8f;

#define B_ 64
#define S_ 8192
#define D_ 128
#define K_ 8
#define SCALE_ 5.0f
#define DECAY_ (1.0f - 1.0f / 8192.0f)
#define EPS_ 1e-30f

// ---------------------------------------------------------------------------
// Kernel 1: per (b, 64-row s-chunk):
//   - stage keys (padded to 16 rows) and a 64x128 memory tile in LDS
//   - fused write memory_w = DECAY*memory (saves a whole 256MB re-read later)
//   - compute key / memory row inverse norms
//   - sim[k][s] = 5 * <key_k, mem_s> * kinv * minv via v_wmma_f32_16x16x4_f32
// 128 threads = 4 waves; wave w handles tile rows [16w, 16w+16).
// ---------------------------------------------------------------------------
__global__ __launch_bounds__(128) void k1_sim(const float* __restrict__ memory,
                                              const float* __restrict__ keys,
                                              float* __restrict__ memw,
                                              float* __restrict__ sim) {
  const int sc = blockIdx.x;   // 0..127  (64 rows each)
  const int b  = blockIdx.y;   // 0..63
  const int t  = threadIdx.x;
  const int s0 = sc * 64;

  __shared__ float tile[64 * 132];   // memory rows, padded stride
  __shared__ float keyl[16 * 132];   // keys, rows 8..15 zero
  __shared__ float norminv[64];
  __shared__ float keyinv[16];

  // stage keys (8x128) + zero-pad rows 8..15
  for (int j = 0; j < 2; ++j) {
    int idx = t + j * 128;
    int r = idx >> 5, c4 = (idx & 31) * 4;
    float4 g = *(const float4*)(keys + ((size_t)b * K_ + r) * D_ + c4);
    *(float4*)&keyl[r * 132 + c4] = g;
    *(float4*)&keyl[(r + 8) * 132 + c4] = make_float4(0.f, 0.f, 0.f, 0.f);
  }
  // stage memory tile + fused decayed memory_w store
  for (int j = 0; j < 16; ++j) {
    int idx = t + j * 128;
    int r = idx >> 5, c4 = (idx & 31) * 4;
    size_t go = ((size_t)b * S_ + s0 + r) * D_ + c4;
    float4 g = *(const float4*)(memory + go);
    *(float4*)&tile[r * 132 + c4] = g;
    float4 wv = make_float4(DECAY_ * g.x, DECAY_ * g.y, DECAY_ * g.z, DECAY_ * g.w);
    *(float4*)(memw + go) = wv;
  }
  __syncthreads();

  if (t < 16) {
    float s = 0.f;
    if (t < 8) {
      for (int i = 0; i < D_; ++i) { float v = keyl[t * 132 + i]; s += v * v; }
    }
    keyinv[t] = (t < 8) ? 1.0f / (EPS_ + sqrtf(s)) : 0.0f;
  }
  if (t < 64) {
    float s = 0.f;
    for (int i = 0; i < D_; ++i) { float v = tile[t * 132 + i]; s += v * v; }
    norminv[t] = 1.0f / (EPS_ + sqrtf(s));
  }
  __syncthreads();

  const int w = t >> 5, lane = t & 31;
  const int m = lane & 15, half = lane >> 4;
  const int rb = w * 16;                     // this wave's 16 memory rows
  v8f c = {};
  for (int d = 0; d < D_; d += 4) {
    // A = keys 16x4 : lane holds A[m][2*half + v]
    v2f a  = *(const v2f*)&keyl[m * 132 + d + half * 2];
    // B = mem^T 4x16: lane holds B[2*half + v][n], n == lane%16
    v2f bb = *(const v2f*)&tile[(rb + m) * 132 + d + half * 2];
    c = __builtin_amdgcn_wmma_f32_16x16x4_f32(false, a, false, bb, (short)0, c,
                                              false, false);
  }
  if (half == 0) {
    // C: vgpr v = row k=v (half 0), col n = lane = s index
    float minv = norminv[rb + m] * SCALE_;
    size_t sabs = (size_t)s0 + rb + m;
    for (int v = 0; v < K_; ++v)
      sim[((size_t)b * K_ + v) * S_ + sabs] = c[v] * keyinv[v] * minv;
  }
}

// ---------------------------------------------------------------------------
// Kernel 2: per (b,k) row: softmax max and 1/sum(exp) over S=8192.
// ---------------------------------------------------------------------------
__global__ __launch_bounds__(256) void k2_stats(const float* __restrict__ sim,
                                                float* __restrict__ rmax,
                                                float* __restrict__ rsinv) {
  const int r = blockIdx.x;                  // 0..511
  const int t = threadIdx.x;
  const float* p = sim + (size_t)r * S_;
  __shared__ float red[256];
  float m = -3.4e38f;
  for (int i = t; i < S_; i += 256) m = fmaxf(m, p[i]);
  red[t] = m; __syncthreads();
  for (int off = 128; off; off >>= 1) {
    if (t < off) red[t] = fmaxf(red[t], red[t + off]);
    __syncthreads();
  }
  float mx = red[0]; __syncthreads();
  float s = 0.f;
  for (int i = t; i < S_; i += 256) s += expf(p[i] - mx);
  red[t] = s; __syncthreads();
  for (int off = 128; off; off >>= 1) {
    if (t < off) red[t] += red[t + off];
    __syncthreads();
  }
  if (t == 0) { rmax[r] = mx; rsinv[r] = 1.0f / red[0]; }
}

// ---------------------------------------------------------------------------
// Kernel 3: per (b, 64-row s-chunk):
//   - stage memory tile TRANSPOSED ([d][s], stride 68) and att tile in LDS
//   - usage_r written straight into usage_w output slot
//   - partial result (attention @ memory) via WMMA; wave w owns d-cols [16w,16w+16)
// 256 threads = 8 waves.
// ---------------------------------------------------------------------------
__global__ __launch_bounds__(256) void k3_attn(const float* __restrict__ memory,
                                               const float* __restrict__ usage,
                                               const float* __restrict__ sim,
                                               const float* __restrict__ rmax,
                                               const float* __restrict__ rsinv,
                                               float* __restrict__ usage_out,
                                               float* __restrict__ part) {
  const int sc = blockIdx.x;   // 0..127
  const int b  = blockIdx.y;
  const int t  = threadIdx.x;
  const int s0 = sc * 64;

  __shared__ float memT[128 * 68];   // [d][s_local], padded stride
  __shared__ float attl[16 * 68];    // [k][s_local], rows 8..15 zero

  for (int j = 0; j < 8; ++j) {
    int idx = t + j * 256;
    int r = idx >> 5, c4 = (idx & 31) * 4;       // r = s_local, c4 = d
    float4 g = *(const float4*)(memory + ((size_t)b * S_ + s0 + r) * D_ + c4);
    memT[(c4 + 0) * 68 + r] = g.x;
    memT[(c4 + 1) * 68 + r] = g.y;
    memT[(c4 + 2) * 68 + r] = g.z;
    memT[(c4 + 3) * 68 + r] = g.w;
  }
  for (int j = 0; j < 2; ++j) {
    int idx = t + j * 256;
    int k = idx >> 6, sl = idx & 63;
    int row = b * K_ + k;
    float av = expf(sim[(size_t)row * S_ + s0 + sl] - rmax[row]) * rsinv[row];
    attl[k * 68 + sl] = av;
    attl[(k + 8) * 68 + sl] = 0.0f;
  }
  __syncthreads();

  if (t < 64) {   // usage_r = usage + sum_k attention
    float s = usage[(size_t)b * S_ + s0 + t];
    for (int k = 0; k < K_; ++k) s += attl[k * 68 + t];
    usage_out[(size_t)b * S_ + s0 + t] = s;
  }

  const int w = t >> 5, lane = t & 31;
  const int m = lane & 15, half = lane >> 4;
  const int d0 = w * 16;
  v8f c = {};
  for (int sl = 0; sl < 64; sl += 4) {
    // A = att 16x4  : A[m][2*half+v]
    v2f a  = *(const v2f*)&attl[m * 68 + sl + half * 2];
    // B = mem 4x16  : B[2*half+v][n] = memory[s][d0+n], n == lane%16
    v2f bb = *(const v2f*)&memT[(d0 + m) * 68 + sl + half * 2];
    c = __builtin_amdgcn_wmma_f32_16x16x4_f32(false, a, false, bb, (short)0, c,
                                              false, false);
  }
  if (half == 0) {
    size_t base = ((size_t)b * 128 + sc) * (K_ * D_);
    for (int v = 0; v < K_; ++v)
      part[base + (size_t)v * D_ + d0 + m] = c[v];
  }
}

// ---------------------------------------------------------------------------
// Kernel 4: fixed-order reduction of 128 partials per batch -> result.
// ---------------------------------------------------------------------------
__global__ __launch_bounds__(256) void k4_reduce(const float* __restrict__ part,
                                                 float* __restrict__ out) {
  const int b = blockIdx.x, t = threadIdx.x;
  for (int i = 0; i < 4; ++i) {
    int j = t + i * 256;                       // 0..1023 = k*128+d
    const float* p = part + (size_t)b * 128 * (K_ * D_) + j;
    float s = 0.f;
    for (int sc = 0; sc < 128; ++sc) s += p[(size_t)sc * (K_ * D_)];
    out[(size_t)b * (K_ * D_) + j] = s;
  }
}

// ---------------------------------------------------------------------------
// Kernel 5: per batch: first-index argmin of usage_r, then fix up
//   usage_w[loc]=1 and memory_w[loc][:] = content.
// ---------------------------------------------------------------------------
__global__ __launch_bounds__(256) void k5_fix(const float* __restrict__ content,
                                              float* __restrict__ usage_out,
                                              float* __restrict__ memw) {
  const int b = blockIdx.x, t = threadIdx.x;
  __shared__ float sval[256];
  __shared__ int   sidx[256];
  __shared__ int   locS;
  const float* u = usage_out + (size_t)b * S_;
  float bv = 3.4e38f; int bi = 0;
  for (int i = t; i < S_; i += 256) {
    float v = u[i];
    if (v < bv) { bv = v; bi = i; }
  }
  sval[t] = bv; sidx[t] = bi; __syncthreads();
  for (int off = 128; off; off >>= 1) {
    if (t < off) {
      float v2 = sval[t + off]; int i2 = sidx[t + off];
      if (v2 < sval[t] || (v2 == sval[t] && i2 < sidx[t])) {
        sval[t] = v2; sidx[t] = i2;
      }
    }
    __syncthreads();
  }
  if (t == 0) {
    locS = sidx[0];
    usage_out[(size_t)b * S_ + locS] = 1.0f;
  }
  __syncthreads();
  int loc = locS;
  if (t < D_)
    memw[((size_t)b * S_ + loc) * D_ + t] = content[(size_t)b * D_ + t];
}

// ---------------------------------------------------------------------------
extern "C" void kernel_launch(void* const* d_in, const int* in_sizes, int n_in,
                              void* d_out, int out_size, void* d_ws, size_t ws_size,
                              hipStream_t stream) {
  (void)in_sizes; (void)n_in; (void)out_size; (void)ws_size;
  const float* memory  = (const float*)d_in[0];
  const float* usage   = (const float*)d_in[1];
  const float* keys    = (const float*)d_in[2];
  const float* content = (const float*)d_in[3];

  float* out     = (float*)d_out;
  float* result  = out;                                   // B*K*D = 65536
  float* memw    = out + (size_t)B_ * K_ * D_;            // B*S*D
  float* usage_w = memw + (size_t)B_ * S_ * D_;           // B*S

  float* ws   = (float*)d_ws;
  float* sim  = ws;                                       // B*K*S   = 4 Mi floats
  float* part = sim + (size_t)B_ * K_ * S_;               // B*128*K*D = 8 Mi floats
  float* rmx  = part + (size_t)B_ * 128 * K_ * D_;        // 512
  float* rsi  = rmx + B_ * K_;                            // 512

  k1_sim  <<<dim3(128, 64), 128, 0, stream>>>(memory, keys, memw, sim);
  k2_stats<<<B_ * K_,       256, 0, stream>>>(sim, rmx, rsi);
  k3_attn <<<dim3(128, 64), 256, 0, stream>>>(memory, usage, sim, rmx, rsi,
                                              usage_w, part);
  k4_reduce<<<B_,           256, 0, stream>>>(part, result);
  k5_fix  <<<B_,            256, 0, stream>>>(content, usage_w, memw);
}